// TimeSeriesSensorPredictor_42296837931437
// MI455X (gfx1250) — compile-verified
//
#include <hip/hip_runtime.h>
#include <math.h>

// Problem constants (match reference): B=4096, T=1024, F=3
#define B_      4096
#define T_      1024
#define F_      3
#define BB      128               // batch rows (chains) per block
#define TC      32                // timesteps per LDS chunk
#define THREADS 256               // 8 wave32s
#define ROWF    (TC * F_)         // 96 dwords of payload per tile row
#define LROW    (ROWF + ROWF/32)  // 99 dwords per row incl. TDM pad (1 dw / 32 dw)
#define CHUNKS  (T_ / TC)         // 32
#define ROWQ    (ROWF / 4)        // 24 float4 per row
#define TILEQ   (BB * ROWQ)       // 3072 float4 per tile

typedef __attribute__((ext_vector_type(2))) float        v2f;
typedef __attribute__((ext_vector_type(8))) float        v8f;
typedef __attribute__((ext_vector_type(4))) unsigned int v4u;
typedef __attribute__((ext_vector_type(4))) int          v4i;
typedef __attribute__((ext_vector_type(8))) int          v8i;

#if defined(__has_builtin)
#if __has_builtin(__builtin_amdgcn_tanhf)
#define TANHF __builtin_amdgcn_tanhf     // gfx1250 V_TANH_F32 (TRANS)
#endif
#if __has_builtin(__builtin_amdgcn_tensor_load_to_lds) && \
    __has_builtin(__builtin_amdgcn_s_wait_tensorcnt)
#define USE_TDM 1
#endif
#endif
#ifndef TANHF
#define TANHF tanhf
#endif

// LDS dword offset for row r, payload dword x — matches TDM pad of 1 dword
// after every 32 stored dwords: off(n) = n + n/32, n = r*ROWF + x.
// Effective row stride 99 (odd mod 64 banks) => conflict-free scan access.
__device__ __forceinline__ int ldsoff(int r, int x) {
    const int n = r * ROWF + x;
    return n + (n >> 5);
}

#ifdef USE_TDM
// Issue one TDM 2D tile load: BB rows x ROWF dwords, row stride T_*F_ dwords.
__device__ __forceinline__ void tdm_load_tile(const float* gsrc,
                                              unsigned int lds_byte) {
    const unsigned long long ga = (unsigned long long)(uintptr_t)gsrc;
    v4u g0;
    g0.x = 1u;                                           // count=1, user mode
    g0.y = lds_byte;                                     // lds_addr (bytes)
    g0.z = (unsigned int)(ga & 0xFFFFFFFFull);           // global_addr[31:0]
    g0.w = (unsigned int)((ga >> 32) & 0x1FFFFFFull)     // global_addr[56:32]
         | (2u << 30);                                   // type = 2 ("image")
    v8i g1;
    g1[0] = (int)((2u << 16)      // data_size = 4 bytes
                | (1u << 20)      // pad_enable
                | (4u << 22));    // pad_interval code 4 = every 32 dwords
                                  // pad_amount = 0 -> 1 dword
    g1[1] = (int)(((unsigned)(T_ * F_) & 0xFFFFu) << 16);          // tensor_dim0 lo16
    g1[2] = (int)((((unsigned)(T_ * F_) >> 16) & 0xFFFFu)          // tensor_dim0 hi16
                | (((unsigned)BB & 0xFFFFu) << 16));               // tensor_dim1 lo16
    g1[3] = (int)(((unsigned)ROWF & 0xFFFFu) << 16);               // tile_dim0
    g1[4] = (int)((unsigned)BB & 0xFFFFu);                         // tile_dim1
    g1[5] = (int)(unsigned)(T_ * F_);                              // dim0_stride lo32
    g1[6] = 0;                                                     // stride hi / dim1_stride
    g1[7] = 0;
    const v4i z4 = {0, 0, 0, 0};
#if __clang_major__ >= 23
    const v8i z8 = {0, 0, 0, 0, 0, 0, 0, 0};
    __builtin_amdgcn_tensor_load_to_lds(g0, g1, z4, z4, z8, 0);
#else
    __builtin_amdgcn_tensor_load_to_lds(g0, g1, z4, z4, 0);
#endif
}
#endif

__global__ __launch_bounds__(THREADS)
void rnn_fused_42296837931437(const float* __restrict__ X,
                              const float* __restrict__ H0,
                              const float* __restrict__ Wih,
                              const float* __restrict__ Whh,
                              const float* __restrict__ bih,
                              const float* __restrict__ bhh,
                              float* __restrict__ out)
{
    __shared__ float lds[2][BB * LROW];   // 2 x 50,688 B = 101,376 B (<320K/WGP)

    const int tid   = threadIdx.x;
    const int b0    = blockIdx.x * BB;
    const int lane  = tid & 31;
    const int wv    = tid >> 5;
    const int l16   = lane & 15;
    const int nhalf = lane >> 4;          // 0: K={0,1}, 1: K={2,pad}

    // ---- wave-invariant WMMA B operand: B[k][n] = Wih[n][k], padded 4x16 ----
    v2f bmat;
    {
        const int n = l16, k0 = nhalf * 2;
        bmat.x = (n < F_) ? Wih[n * F_ + k0] : 0.0f;
        bmat.y = (n < F_ && nhalf == 0) ? Wih[n * F_ + 1] : 0.0f;
    }
    const v8f czero = {0.f, 0.f, 0.f, 0.f, 0.f, 0.f, 0.f, 0.f};

    // ---- recurrent weights + fused bias (b_ih + b_hh) in registers ----
    const float w00 = Whh[0], w01 = Whh[1], w02 = Whh[2];
    const float w10 = Whh[3], w11 = Whh[4], w12 = Whh[5];
    const float w20 = Whh[6], w21 = Whh[7], w22 = Whh[8];
    const float bs0 = bih[0] + bhh[0];
    const float bs1 = bih[1] + bhh[1];
    const float bs2 = bih[2] + bhh[2];

    float h0 = 0.f, h1 = 0.f, h2 = 0.f;
    if (tid < BB) {
        const float* hp = H0 + (size_t)(b0 + tid) * F_;   // H is [1,B,F]
        h0 = hp[0]; h1 = hp[1]; h2 = hp[2];
    }

    const float* xbase = X + (size_t)b0 * (T_ * F_);

#ifdef USE_TDM
    if (wv == 0) {   // prologue: prefetch chunks 0 and 1 via the Tensor Data Mover
        tdm_load_tile(xbase + 0 * (size_t)ROWF, (unsigned)(uintptr_t)&lds[0][0]);
        tdm_load_tile(xbase + 1 * (size_t)ROWF, (unsigned)(uintptr_t)&lds[1][0]);
    }
#endif

    for (int c = 0; c < CHUNKS; ++c) {
        float* buf = lds[c & 1];

#ifdef USE_TDM
        if (wv == 0) {
            // TENSORcnt completes in-order: <=1 outstanding => chunk c landed.
            if (c < CHUNKS - 1) __builtin_amdgcn_s_wait_tensorcnt(1);
            else                __builtin_amdgcn_s_wait_tensorcnt(0);
        }
        __syncthreads();
#else
        // -------- synchronous staging fallback, coalesced float4 --------
        for (int i = tid; i < TILEQ; i += THREADS) {
            const int r = i / ROWQ, q = i % ROWQ;
            const float4 v = *(const float4*)(xbase + (size_t)r * (T_ * F_)
                                              + (size_t)c * ROWF + q * 4);
            const int dd = ldsoff(r, q * 4);   // 4q..4q+3 never straddle a pad
            buf[dd + 0] = v.x; buf[dd + 1] = v.y;
            buf[dd + 2] = v.z; buf[dd + 3] = v.w;
        }
        __syncthreads();
#endif

        // ---------- in-place input projection via V_WMMA_F32_16X16X4_F32 ----
        for (int g = wv; g < (BB * TC) / 16; g += THREADS / 32) {
            const int r0 = g * 16;
            const int m  = r0 + l16;
            const int bl = m >> 5, s = m & (TC - 1);
            v2f amat;                                    // ISA 16x4 A layout
            amat.x = buf[ldsoff(bl, 3 * s + nhalf * 2)];
            amat.y = (nhalf == 0) ? buf[ldsoff(bl, 3 * s + 1)] : 0.0f;
            v8f acc = __builtin_amdgcn_wmma_f32_16x16x4_f32(
                false, amat, false, bmat, (short)0, czero, false, false);
            if (l16 < F_) {                              // scatter N<3 columns
                #pragma unroll
                for (int j = 0; j < 8; ++j) {            // D: M=j / M=8+j
                    const int mm  = r0 + j + nhalf * 8;
                    const int bl2 = mm >> 5, s2 = mm & (TC - 1);
                    buf[ldsoff(bl2, 3 * s2 + l16)] = acc[j];
                }
            }
        }
        __syncthreads();

        // ---------- sequential recurrence, one thread per chain ----------
        if (tid < BB) {
            #pragma unroll 4
            for (int s = 0; s < TC; ++s) {
                const int a0 = ldsoff(tid, 3 * s + 0);   // computed separately:
                const int a1 = ldsoff(tid, 3 * s + 1);   // triples may straddle
                const int a2 = ldsoff(tid, 3 * s + 2);   // a pad boundary
                const float p0 = buf[a0] + bs0 + w00 * h0 + w01 * h1 + w02 * h2;
                const float p1 = buf[a1] + bs1 + w10 * h0 + w11 * h1 + w12 * h2;
                const float p2 = buf[a2] + bs2 + w20 * h0 + w21 * h1 + w22 * h2;
                h0 = TANHF(p0); h1 = TANHF(p1); h2 = TANHF(p2);
                buf[a0] = h0; buf[a1] = h1; buf[a2] = h2;
            }
        }
        __syncthreads();

        // ---------- store tile -> out, coalesced float4 ----------
        for (int i = tid; i < TILEQ; i += THREADS) {
            const int r = i / ROWQ, q = i % ROWQ;
            const int dd = ldsoff(r, q * 4);
            float4 v;
            v.x = buf[dd + 0]; v.y = buf[dd + 1];
            v.z = buf[dd + 2]; v.w = buf[dd + 3];
            *(float4*)(out + (size_t)(b0 + r) * (T_ * F_)
                           + (size_t)c * ROWF + q * 4) = v;
        }
        __syncthreads();   // all LDS reads of buf done -> safe to refill

#ifdef USE_TDM
        if (wv == 0 && c + 2 < CHUNKS)   // refill just-consumed buffer
            tdm_load_tile(xbase + (size_t)(c + 2) * ROWF,
                          (unsigned)(uintptr_t)&lds[c & 1][0]);
#endif
    }

    // ---------- h_n tail: out[B*T*F ..] = last hidden state ----------
    if (tid < BB) {
        float* hn = out + (size_t)B_ * T_ * F_ + (size_t)(b0 + tid) * F_;
        hn[0] = h0; hn[1] = h1; hn[2] = h2;
    }
}

extern "C" void kernel_launch(void* const* d_in, const int* in_sizes, int n_in,
                              void* d_out, int out_size, void* d_ws, size_t ws_size,
                              hipStream_t stream) {
    (void)in_sizes; (void)n_in; (void)out_size; (void)d_ws; (void)ws_size;
    const float* X   = (const float*)d_in[0];
    const float* H0  = (const float*)d_in[1];
    const float* Wih = (const float*)d_in[2];
    const float* Whh = (const float*)d_in[3];
    const float* bih = (const float*)d_in[4];
    const float* bhh = (const float*)d_in[5];
    float* out = (float*)d_out;
    rnn_fused_42296837931437<<<B_ / BB, THREADS, 0, stream>>>(
        X, H0, Wih, Whh, bih, bhh, out);
}